// Graph_split_projection_17824114278488
// MI455X (gfx1250) — compile-verified
//
#include <hip/hip_runtime.h>
#include <math.h>

// Problem constants (reference: B,C,H,W = 4,256,64,64; D=256, K=32)
#define BATCH 4
#define CDIM  256      // C == D
#define NPIX  4096     // H*W
#define KCL   32       // K clusters
#define GRP   8        // C / K
#define NSEG  8        // split-K segments for the nlen=4096 GEMMs
#define SEGLEN (NPIX / NSEG)

typedef float v2f __attribute__((ext_vector_type(2)));
typedef float v8f __attribute__((ext_vector_type(8)));

__device__ __forceinline__ v8f wmma_f32(v2f a, v2f b, v8f c) {
    // V_WMMA_F32_16X16X4_F32 : D = A(16x4) * B(4x16) + C(16x16)
    return __builtin_amdgcn_wmma_f32_16x16x4_f32(
        /*neg_a=*/false, a, /*neg_b=*/false, b,
        /*c_mod=*/(short)0, c, /*reuse_a=*/false, /*reuse_b=*/false);
}

// ---------------------------------------------------------------------------
// Kernel 1: grouped 1x1 conv -> softmax over pixels (axis=N) per (b,k).
// Also emits s2 = scale*scale (each of the 128 blocks writes 64 entries).
// grid = B*K = 128 blocks, 256 threads. Each thread owns 16 pixels.
// ---------------------------------------------------------------------------
__global__ __launch_bounds__(256)
void phi_softmax_kernel(const float* __restrict__ x,
                        const float* __restrict__ w_phi,
                        const float* __restrict__ scale,
                        float* __restrict__ phi,      // [B][N][K]
                        float* __restrict__ s2out) {  // [D][K]
    const int bk = blockIdx.x;
    const int b = bk >> 5, k = bk & 31;
    const int tid = threadIdx.x;
    __shared__ float red[256];

    float wp[GRP];
#pragma unroll
    for (int c = 0; c < GRP; ++c) wp[c] = w_phi[k * GRP + c];

    const float* xb = x + ((long)b * CDIM + k * GRP) * NPIX;

    float lg[16], e[16];
    float mymax = -INFINITY;
#pragma unroll
    for (int i = 0; i < 16; ++i) {
        const int n = tid + (i << 8);
        float s = 0.f;
#pragma unroll
        for (int c = 0; c < GRP; ++c) s = fmaf(wp[c], xb[c * NPIX + n], s);
        lg[i] = s;
        mymax = fmaxf(mymax, s);
    }
    red[tid] = mymax; __syncthreads();
    for (int o = 128; o > 0; o >>= 1) {
        if (tid < o) red[tid] = fmaxf(red[tid], red[tid + o]);
        __syncthreads();
    }
    const float bmax = red[0]; __syncthreads();

    float mysum = 0.f;
#pragma unroll
    for (int i = 0; i < 16; ++i) { e[i] = __expf(lg[i] - bmax); mysum += e[i]; }
    red[tid] = mysum; __syncthreads();
    for (int o = 128; o > 0; o >>= 1) {
        if (tid < o) red[tid] += red[tid + o];
        __syncthreads();
    }
    const float inv = 1.f / red[0];

#pragma unroll
    for (int i = 0; i < 16; ++i) {
        const int n = tid + (i << 8);
        phi[((long)b * NPIX + n) * KCL + k] = e[i] * inv;
    }

    // s2 = scale^2 : 8192 entries spread over 128 blocks (64 each)
    if (tid < 64) {
        const int idx = bk * 64 + tid;
        const float sc = scale[idx];
        s2out[idx] = sc * sc;
    }
}

// ---------------------------------------------------------------------------
// Batched WMMA-f32 GEMM with split-K:
//   Out[seg][b][m][k'] = sum_{n in segment} A[b][m][n] * Bm[b][n][k']
// M = 256 rows, K' = 32 cols, segment length seglen (multiple of 8).
// 128 waves per segment (B*16*2 tiles); nseg segments -> grid = 32*nseg
// blocks of 128 threads. Two interleaved accumulators break the WMMA RAW
// chain; single bumped pointers + constant indices let the loads use the
// 24-bit immediate offset field; prefetch (locality 3 -> WGP scope) streams
// the strided B rows ahead of the consuming loads.
// ---------------------------------------------------------------------------
__global__ __launch_bounds__(128)
void wmma_gemm_nk(const float* __restrict__ A, long aBatchStride, int lda,
                  const float* __restrict__ Bm, long bBatchStride,
                  float* __restrict__ Out, long oBatchStride, long segStride,
                  int seglen) {
    const int wave = threadIdx.x >> 5;
    const int lane = threadIdx.x & 31;
    const int wid = blockIdx.x * 4 + wave;
    const int seg = wid >> 7;                      // 128 waves per segment
    const int t = wid & 127;
    const int b = t >> 5;
    const int rem = t & 31;
    const int m0 = (rem >> 1) << 4;                // 0..240
    const int k0 = (rem & 1) << 4;                 // 0 or 16
    const int lmod = lane & 15, lhalf = lane >> 4;

    const float* Ab = A + (long)b * aBatchStride;
    const float* Bb = Bm + (long)b * bBatchStride;
    float* Ob = Out + (long)seg * segStride + (long)b * oBatchStride;

    const int nbeg = seg * seglen;
    // A frag: element (m0+lmod, n + 2*lhalf + v)  -> contiguous float pair
    const float* aptr = Ab + (long)(m0 + lmod) * lda + 2 * lhalf + nbeg;
    // B frag: element (n + 2*lhalf + v, k0+lmod)
    const float* bptr = Bb + (long)(nbeg + 2 * lhalf) * KCL + k0 + lmod;

    v8f acc0 = {0.f, 0.f, 0.f, 0.f, 0.f, 0.f, 0.f, 0.f};
    v8f acc1 = acc0;
    const int iters = seglen >> 3;
#pragma unroll 2
    for (int it = 0; it < iters; ++it) {
        v2f af0; af0.x = aptr[0]; af0.y = aptr[1];
        v2f af1; af1.x = aptr[4]; af1.y = aptr[5];
        v2f bf0; bf0.x = bptr[0 * KCL]; bf0.y = bptr[1 * KCL];
        v2f bf1; bf1.x = bptr[4 * KCL]; bf1.y = bptr[5 * KCL];
        // stream the strided B rows ahead (speculative; OOB silently dropped)
        __builtin_prefetch(bptr + 32 * KCL, 0, 3);
        acc0 = wmma_f32(af0, bf0, acc0);
        acc1 = wmma_f32(af1, bf1, acc1);
        aptr += 8;
        bptr += 8 * KCL;
    }
    const v8f acc = acc0 + acc1;

    // D layout: M = v + 8*lhalf, N = lmod
    float* orow = Ob + (long)(m0 + 8 * lhalf) * KCL + k0 + lmod;
#pragma unroll
    for (int v = 0; v < 8; ++v) orow[v * KCL] = acc[v];
}

// ---------------------------------------------------------------------------
// Sum split-K partials: Out[i] = sum_seg P[seg][i], i < B*D*K (=32768).
// Fixed ascending seg order -> bit-deterministic. grid 128 x 256.
// ---------------------------------------------------------------------------
__global__ __launch_bounds__(256)
void reduce_partials_kernel(const float* __restrict__ P, float* __restrict__ Out) {
    const int i = blockIdx.x * 256 + threadIdx.x;
    float s = 0.f;
#pragma unroll
    for (int g = 0; g < NSEG; ++g) s += P[(long)g * (BATCH * CDIM * KCL) + i];
    Out[i] = s;
}

// ---------------------------------------------------------------------------
// code normalization (over D) + s2*code + t3[b,k] = sum_d s2*code^2
// grid = B*K = 128 blocks, 256 threads (thread = d)
// ---------------------------------------------------------------------------
__global__ __launch_bounds__(256)
void code_norm_kernel(const float* __restrict__ code_raw,
                      const float* __restrict__ s2,
                      float* __restrict__ code,
                      float* __restrict__ s2code,
                      float* __restrict__ t3) {
    const int bk = blockIdx.x;
    const int b = bk >> 5, k = bk & 31;
    const int d = threadIdx.x;
    __shared__ float red[256];

    const long idx = ((long)b * CDIM + d) * KCL + k;
    const float cr = code_raw[idx];
    red[d] = cr * cr; __syncthreads();
    for (int o = 128; o > 0; o >>= 1) {
        if (d < o) red[d] += red[d + o];
        __syncthreads();
    }
    const float nrm = sqrtf(red[0]); __syncthreads();
    const float c = cr / fmaxf(nrm, 1e-12f);
    const float s2v = s2[d * KCL + k];
    const float sc = s2v * c;
    code[idx] = c;
    s2code[idx] = sc;

    red[d] = sc * c; __syncthreads();
    for (int o = 128; o > 0; o >>= 1) {
        if (d < o) red[d] += red[d + o];
        __syncthreads();
    }
    if (d == 0) t3[bk] = red[0];
}

// ---------------------------------------------------------------------------
// Fused distance + softmax(K) kernel.
//   dist[n,k] = sum_d x^2*s2 - 2*sum_d x*(s2*code) + t3[k]
//   Q = softmax(-0.5*dist) over k (32)  -> written straight to d_out.
// 4 independent WMMA accumulators per wave: {t1,t2} x {k-tile 0, k-tile 1}.
// grid = 256 blocks x 128 threads (4 waves) -> 1024 waves = B * N/16 tiles.
// ---------------------------------------------------------------------------
__global__ __launch_bounds__(128)
void dist_q_kernel(const float* __restrict__ x,      // [B][D][N]
                   const float* __restrict__ s2,     // [D][K]
                   const float* __restrict__ s2code, // [B][D][K]
                   const float* __restrict__ t3,     // [B][K]
                   float* __restrict__ Qout) {       // [B][N][K]
    const int wave = threadIdx.x >> 5;
    const int lane = threadIdx.x & 31;
    const int tile = blockIdx.x * 4 + wave;        // 0..1023
    const int b = tile >> 8;
    const int n0 = (tile & 255) << 4;
    const int lmod = lane & 15, lhalf = lane >> 4;

    // per-lane streams, bumped each d-step; constant indices -> imm offsets
    const float* xptr = x + (long)b * CDIM * NPIX + (long)(2 * lhalf) * NPIX + n0 + lmod;
    const float* p2  = s2 + (2 * lhalf) * KCL + lmod;
    const float* p2c = s2code + (long)b * CDIM * KCL + (2 * lhalf) * KCL + lmod;

    v8f acc10 = {0.f,0.f,0.f,0.f,0.f,0.f,0.f,0.f};
    v8f acc11 = acc10, acc20 = acc10, acc21 = acc10;

#pragma unroll 4
    for (int it = 0; it < CDIM / 4; ++it) {
        const float a0 = xptr[0];
        const float a1 = xptr[NPIX];
        // each d-step touches new 16KB-strided lines of x: pull them early
        __builtin_prefetch(xptr + 16 * NPIX, 0, 3);
        v2f af  = {a0, a1};
        v2f af2 = {a0 * a0, a1 * a1};
        v2f b10 = { p2[0],        p2[KCL] };
        v2f b11 = { p2[16],       p2[KCL + 16] };
        v2f b20 = { p2c[0],       p2c[KCL] };
        v2f b21 = { p2c[16],      p2c[KCL + 16] };
        acc10 = wmma_f32(af2, b10, acc10);
        acc11 = wmma_f32(af2, b11, acc11);
        acc20 = wmma_f32(af,  b20, acc20);
        acc21 = wmma_f32(af,  b21, acc21);
        xptr += 4 * NPIX;
        p2  += 4 * KCL;
        p2c += 4 * KCL;
    }

    const float t3a = t3[b * KCL + lmod];
    const float t3b = t3[b * KCL + 16 + lmod];

#pragma unroll
    for (int v = 0; v < 8; ++v) {
        // lane holds (n = n0+v+8*lhalf, k = lmod) and (.., k = 16+lmod)
        float s0 = -0.5f * (acc10[v] - 2.f * acc20[v] + t3a);
        float s1 = -0.5f * (acc11[v] - 2.f * acc21[v] + t3b);
        float m = fmaxf(s0, s1);
        for (int off = 1; off < 16; off <<= 1) m = fmaxf(m, __shfl_xor(m, off));
        const float e0 = __expf(s0 - m);
        const float e1 = __expf(s1 - m);
        float s = e0 + e1;
        for (int off = 1; off < 16; off <<= 1) s += __shfl_xor(s, off);
        const float inv = 1.f / s;
        const int n = n0 + v + 8 * lhalf;
        float* qrow = Qout + ((long)b * NPIX + n) * KCL;
        qrow[lmod]      = e0 * inv;
        qrow[16 + lmod] = e1 * inv;
    }
}

// ---------------------------------------------------------------------------
// Qsum[b,k] = sum_n Q[b,n,k]   (deterministic tree reduce, no fp atomics)
// grid = B*K = 128 blocks, 256 threads
// ---------------------------------------------------------------------------
__global__ __launch_bounds__(256)
void qsum_kernel(const float* __restrict__ Q, float* __restrict__ Qsum) {
    const int bk = blockIdx.x;
    const int b = bk >> 5, k = bk & 31;
    const int tid = threadIdx.x;
    __shared__ float red[256];
    const float* qb = Q + (long)b * NPIX * KCL + k;
    float s = 0.f;
#pragma unroll
    for (int i = 0; i < 16; ++i) s += qb[(long)(tid + (i << 8)) * KCL];
    red[tid] = s; __syncthreads();
    for (int o = 128; o > 0; o >>= 1) {
        if (tid < o) red[tid] += red[tid + o];
        __syncthreads();
    }
    if (tid == 0) Qsum[bk] = red[0];
}

// ---------------------------------------------------------------------------
// Z = normalize_d( scale * (M2/Qsum - code) )   (plain sqrt, no eps)
// grid = B*K = 128 blocks, 256 threads (thread = d)
// ---------------------------------------------------------------------------
__global__ __launch_bounds__(256)
void z_kernel(const float* __restrict__ M2,
              const float* __restrict__ Qsum,
              const float* __restrict__ code,
              const float* __restrict__ scale,
              float* __restrict__ Z) {
    const int bk = blockIdx.x;
    const int b = bk >> 5, k = bk & 31;
    const int d = threadIdx.x;
    __shared__ float red[256];
    const long idx = ((long)b * CDIM + d) * KCL + k;
    const float z_ = scale[d * KCL + k] * (M2[idx] / Qsum[bk] - code[idx]);
    red[d] = z_ * z_; __syncthreads();
    for (int o = 128; o > 0; o >>= 1) {
        if (d < o) red[d] += red[d + o];
        __syncthreads();
    }
    Z[idx] = z_ * rsqrtf(red[0]);
}

// ---------------------------------------------------------------------------
extern "C" void kernel_launch(void* const* d_in, const int* in_sizes, int n_in,
                              void* d_out, int out_size, void* d_ws, size_t ws_size,
                              hipStream_t stream) {
    const float* x       = (const float*)d_in[0];  // B,C,H,W
    const float* w_theta = (const float*)d_in[1];  // D,C
    const float* w_phi   = (const float*)d_in[2];  // K,C/K
    const float* scale   = (const float*)d_in[3];  // D,K

    float* Z = (float*)d_out;                      // B*D*K = 32768
    float* Q = (float*)d_out + BATCH * CDIM * KCL; // B*N*K = 524288

    const long CK = (long)BATCH * CDIM * KCL;      // 32768
    float* ws       = (float*)d_ws;
    float* phi      = ws;                              // 524288
    float* M1       = phi + (long)BATCH * NPIX * KCL;  // 32768  (x @ phi)
    float* code_raw = M1 + CK;                         // 32768
    float* code     = code_raw + CK;                   // 32768
    float* s2code   = code + CK;                       // 32768
    float* t3       = s2code + CK;                     // 128
    float* Qsum     = t3 + BATCH * KCL;                // 128
    float* M2       = Qsum + BATCH * KCL;              // 32768  (x @ Q)
    float* s2       = M2 + CK;                         // 8192
    float* partials = s2 + (long)CDIM * KCL;           // NSEG*32768 = 262144
    // total ~3.8 MB of ws

    // 1) grouped proj + softmax over N (also emits s2 = scale^2)
    phi_softmax_kernel<<<BATCH * KCL, 256, 0, stream>>>(x, w_phi, scale, phi, s2);
    // 2) M1 = x @ phi  (split-K over N: 1024 waves, then deterministic reduce)
    wmma_gemm_nk<<<32 * NSEG, 128, 0, stream>>>(x, (long)CDIM * NPIX, NPIX,
                                                phi, (long)NPIX * KCL,
                                                partials, (long)CDIM * KCL, CK,
                                                SEGLEN);
    reduce_partials_kernel<<<128, 256, 0, stream>>>(partials, M1);
    // 3) code_raw = w_theta @ M1  (single segment, contraction = 256)
    wmma_gemm_nk<<<32, 128, 0, stream>>>(w_theta, 0L, CDIM,
                                         M1, (long)CDIM * KCL,
                                         code_raw, (long)CDIM * KCL, 0L,
                                         CDIM);
    // 4) normalize code over D; s2*code; t3
    code_norm_kernel<<<BATCH * KCL, 256, 0, stream>>>(code_raw, s2, code, s2code, t3);
    // 5) fused distance GEMMs + softmax over K -> Q (direct to output)
    dist_q_kernel<<<256, 128, 0, stream>>>(x, s2, s2code, t3, Q);
    // 6) Qsum
    qsum_kernel<<<BATCH * KCL, 256, 0, stream>>>(Q, Qsum);
    // 7) M2 = x @ Q  (split-K + reduce)
    wmma_gemm_nk<<<32 * NSEG, 128, 0, stream>>>(x, (long)CDIM * NPIX, NPIX,
                                                Q, (long)NPIX * KCL,
                                                partials, (long)CDIM * KCL, CK,
                                                SEGLEN);
    reduce_partials_kernel<<<128, 256, 0, stream>>>(partials, M2);
    // 8) Z epilogue
    z_kernel<<<BATCH * KCL, 256, 0, stream>>>(M2, Qsum, code, scale, Z);
}